// LS2T_54365696032799
// MI455X (gfx1250) — compile-verified
//
#include <hip/hip_runtime.h>
#include <stdint.h>

// Problem constants from the reference
#define B_      32
#define T_      2048
#define F_      128
#define M_      64
#define C_      10          // NUM_LEVELS*(NUM_LEVELS+1)/2
#define NTILE   16          // m-columns per block
#define TCHUNK  64          // timesteps staged per iteration
#define PITCH   132         // LDS pitch: 132%64=4 -> conflict-free strided reads; 528B rows keep 16B align

typedef __attribute__((ext_vector_type(2))) float v2f;
typedef __attribute__((ext_vector_type(8))) float v8f;

__global__ __launch_bounds__(128)
void ls2t_fused_kernel(const float* __restrict__ X,
                       const float* __restrict__ W,
                       const float* __restrict__ bias,
                       float* __restrict__ out)
{
    __shared__ float Wlds[C_][NTILE][PITCH];   // 84.5 KB  weights [c][n][f] (transposed for b64 B reads)
    __shared__ float Xlds[TCHUNK][PITCH];      // 33.8 KB  X chunk [t][f]
    __shared__ float Mlds[C_][TCHUNK][NTILE];  // 40 KB    per-chunk projections
    __shared__ float blds[C_][NTILE];

    const int mt   = blockIdx.x;     // m tile 0..3
    const int b    = blockIdx.y;     // batch 0..31
    const int tid  = threadIdx.x;    // 0..127
    const int wave = tid >> 5;       // 0..3 (wave32)
    const int lane = tid & 31;
    const int half = lane >> 4;      // K half: lanes 0-15 hold K=0,1 ; lanes 16-31 hold K=2,3
    const int ln   = lane & 15;

    // ---- stage kernel slice (transposed) + bias for this m-tile ----
    for (int i = tid; i < C_ * F_ * NTILE; i += 128) {
        int c = i / (F_ * NTILE);
        int r = i % (F_ * NTILE);
        int f = r / NTILE;
        int n = r % NTILE;           // consecutive tid -> consecutive n: coalesced global read
        Wlds[c][n][f] = W[(c * F_ + f) * M_ + mt * NTILE + n];
    }
    if (tid < C_ * NTILE) {
        int c = tid / NTILE, n = tid % NTILE;
        blds[c][n] = bias[c * M_ + mt * NTILE + n];
    }

    // ---- per-lane scan state (lanes 0..15 carry the recurrence) ----
    float y1 = 0.f, y2 = 0.f, y3 = 0.f, y4 = 0.f;
    float s21 = 0.f;
    float s31 = 0.f, s32 = 0.f;
    float s41 = 0.f, s42 = 0.f, s43 = 0.f;

    const float* Xb = X + (size_t)b * T_ * F_;

    for (int t0 = 0; t0 < T_; t0 += TCHUNK) {
        // ---- async-stage X[b, t0:t0+64, :] directly into LDS (ASYNCcnt path) ----
        {
            for (int i = tid; i < TCHUNK * (F_ / 4); i += 128) {
                int tl = i >> 5;                 // row (32 x 16B per row)
                int q  = i & 31;
                const float* src = Xb + (size_t)(t0 + tl) * F_ + q * 4;
                unsigned dst = (unsigned)(uintptr_t)&Xlds[tl][q * 4];   // 16B-aligned LDS offset
                asm volatile("global_load_async_to_lds_b128 %0, %1, off"
                             :: "v"(dst), "v"(src) : "memory");
            }
            asm volatile("s_wait_asynccnt 0x0" ::: "memory");
            if (t0 + TCHUNK < T_)
                __builtin_prefetch(Xb + (size_t)(t0 + TCHUNK) * F_ + tid * 64, 0, 0);
        }
        __syncthreads();

        // ---- GEMM chunk: for each owned component, keep B in VGPRs, run dual-acc row tiles ----
        for (int c = wave; c < C_; c += 4) {      // wave-uniform -> EXEC all-ones for WMMA
            // B operands for this component: 32 x v2f held in registers, one ds_load_b64 each
            v2f Breg[F_ / 4];
            #pragma unroll
            for (int ks = 0; ks < F_ / 4; ++ks) {
                const int f0 = 4 * ks + 2 * half;
                Breg[ks] = *(const v2f*)&Wlds[c][ln][f0];
            }
            const float bv = blds[c][ln];

            #pragma unroll
            for (int rtp = 0; rtp < 2; ++rtp) {   // two row-tile pairs: (0,1) then (2,3)
                const int t0a = (2 * rtp + 0) * 16;
                const int t0b = (2 * rtp + 1) * 16;
                v8f acc0 = {bv, bv, bv, bv, bv, bv, bv, bv};
                v8f acc1 = {bv, bv, bv, bv, bv, bv, bv, bv};

                #pragma unroll
                for (int ks = 0; ks < F_ / 4; ++ks) {
                    const int f0 = 4 * ks + 2 * half;
                    v2f a0 = *(const v2f*)&Xlds[t0a + ln][f0];
                    v2f a1 = *(const v2f*)&Xlds[t0b + ln][f0];
                    acc0 = __builtin_amdgcn_wmma_f32_16x16x4_f32(
                               false, a0, false, Breg[ks], (short)0, acc0, false, false);
                    acc1 = __builtin_amdgcn_wmma_f32_16x16x4_f32(
                               false, a1, false, Breg[ks], (short)0, acc1, false, false);
                }
                // D layout: lane<16 -> rows 0..7 col ln ; lane>=16 -> rows 8..15 col ln
                #pragma unroll
                for (int r = 0; r < 8; ++r) {
                    Mlds[c][t0a + 8 * half + r][ln] = acc0[r];
                    Mlds[c][t0b + 8 * half + r][ln] = acc1[r];
                }
            }
        }
        __syncthreads();

        // ---- sequential iterated-sum scan over this chunk (16 lanes) ----
        if (tid < NTILE) {
            const int n = tid;
            for (int t = 0; t < TCHUNK; ++t) {
                const float m0 = Mlds[0][t][n], m1 = Mlds[1][t][n];
                const float m2 = Mlds[2][t][n], m3 = Mlds[3][t][n];
                const float m4 = Mlds[4][t][n], m5 = Mlds[5][t][n];
                const float m6 = Mlds[6][t][n], m7 = Mlds[7][t][n];
                const float m8 = Mlds[8][t][n], m9 = Mlds[9][t][n];

                y1 += m0;                 // level 1
                y2 += m2 * s21;           // level 2
                const float r2 = m4 * s31;          // level 3
                const float r3 = m5 * s32;
                y3 += r3;
                const float q2 = m7 * s41;          // level 4
                const float q3 = m8 * s42;
                const float q4 = m9 * s43;
                y4 += q4;
                // exclusive-prefix state updates (after use)
                s21 += m1;
                s32 += r2; s31 += m3;
                s43 += q3; s42 += q2; s41 += m6;
            }
        }
        __syncthreads();
    }

    if (tid < NTILE) {
        const int m = mt * NTILE + tid;
        out[(b * 4 + 0) * M_ + m] = y1;
        out[(b * 4 + 1) * M_ + m] = y2;
        out[(b * 4 + 2) * M_ + m] = y3;
        out[(b * 4 + 3) * M_ + m] = y4;
    }
}

extern "C" void kernel_launch(void* const* d_in, const int* in_sizes, int n_in,
                              void* d_out, int out_size, void* d_ws, size_t ws_size,
                              hipStream_t stream) {
    const float* X    = (const float*)d_in[0];   // [32, 2048, 128] f32
    const float* W    = (const float*)d_in[1];   // [10, 128, 64]   f32
    const float* bias = (const float*)d_in[2];   // [10, 64]        f32
    float* out        = (float*)d_out;           // [32, 4, 64]     f32

    dim3 grid(M_ / NTILE, B_);   // 4 x 32 = 128 blocks
    ls2t_fused_kernel<<<grid, 128, 0, stream>>>(X, W, bias, out);
}